// AlgebraicLinear_28527172780630
// MI455X (gfx1250) — compile-verified
//
#include <hip/hip_runtime.h>

typedef __attribute__((ext_vector_type(2))) float v2f;
typedef __attribute__((ext_vector_type(4))) float v4f;
typedef __attribute__((ext_vector_type(8))) float v8f;

constexpr int CIN    = 64;
constexpr int COUT   = 64;
constexpr int NTOK   = 1024 * 2048;   // B * S
constexpr int NTILES = NTOK / 16;     // 16 tokens per wave-tile

// One wave32 computes a 16-token x 64-out tile:
//   D(16x16) += A(16x4) * B(4x16), 16 k-chunks x 4 out-chunks = 64 WMMAs.
// A = x rows (tokens), B = W^T chunks (held resident in VGPRs).
__global__ __launch_bounds__(256) void lin64_wmma_f32(
    const float* __restrict__ x, const float* __restrict__ w,
    const float* __restrict__ bias, float* __restrict__ y) {
  __shared__ __align__(16) float stage[8][16 * COUT];  // 4 KB per wave

  const int lane = threadIdx.x & 31;
  const int wid  = threadIdx.x >> 5;   // wave in block (8 waves)
  const int lo   = lane & 15;          // M / N index within fragment
  const int hi   = lane >> 4;          // selects K pair {0,1} vs {2,3}

  // ---- Preload W^T fragments (B-matrix layout), resident in VGPRs.
  // wf[oc][kc].{x,y} = W[oc*16+lo][kc*4 + 2*hi + {0,1}]
  v2f wf[4][16];
  {
    const float* wp = w + lo * CIN + 2 * hi;
#pragma unroll
    for (int oc = 0; oc < 4; ++oc)
#pragma unroll
      for (int kc = 0; kc < 16; ++kc)
        wf[oc][kc] = *(const v2f*)(wp + oc * 16 * CIN + kc * 4);
  }

  // Per-lane bias: all 8 acc VGPRs of a lane share out-channel oc*16+lo.
  float bv[4];
#pragma unroll
  for (int oc = 0; oc < 4; ++oc) bv[oc] = bias[oc * 16 + lo];

  const int nwaves = (int)(gridDim.x * blockDim.x) >> 5;
  const int gw     = (int)(blockIdx.x * blockDim.x + threadIdx.x) >> 5;

  for (int t = gw; t < NTILES; t += nwaves) {
    // ---- A fragments: af[kc].{x,y} = x[t*16+lo][kc*4 + 2*hi + {0,1}]
    // Streaming data: non-temporal (no reuse, don't thrash L2).
    const float* xp = x + (size_t)(t * 16 + lo) * CIN + 2 * hi;
    v2f af[16];
#pragma unroll
    for (int kc = 0; kc < 16; ++kc)
      af[kc] = __builtin_nontemporal_load((const v2f*)(xp + kc * 4));

    // ---- 64 x V_WMMA_F32_16X16X4_F32, bias folded into C init.
    v8f acc[4];
#pragma unroll
    for (int oc = 0; oc < 4; ++oc) {
      const float b = bv[oc];
      v8f c = {b, b, b, b, b, b, b, b};
#pragma unroll
      for (int kc = 0; kc < 16; ++kc)
        c = __builtin_amdgcn_wmma_f32_16x16x4_f32(
            false, af[kc], false, wf[oc][kc], (short)0, c, false, false);
      acc[oc] = c;
    }

    // ---- Repack D through LDS so global writes are contiguous b128.
    // D layout: acc[oc][r] = Y[t*16 + r + 8*hi][oc*16 + lo]
    float* st = &stage[wid][0];
#pragma unroll
    for (int oc = 0; oc < 4; ++oc)
#pragma unroll
      for (int r = 0; r < 8; ++r)
        st[(r + 8 * hi) * COUT + oc * 16 + lo] = acc[oc][r];

    // LDS is in-order per wave; staging slice is wave-private -> no barrier,
    // just wait for the DS stores before re-reading.
    asm volatile("s_wait_dscnt 0" ::: "memory");

    float* yp = y + (size_t)t * 16 * COUT;
#pragma unroll
    for (int i = 0; i < 8; ++i) {
      v4f v = *(const v4f*)(st + (i * 32 + lane) * 4);
      __builtin_nontemporal_store(v, (v4f*)(yp + (i * 32 + lane) * 4));
    }
  }
}

extern "C" void kernel_launch(void* const* d_in, const int* in_sizes, int n_in,
                              void* d_out, int out_size, void* d_ws, size_t ws_size,
                              hipStream_t stream) {
  const float* x = (const float*)d_in[0];  // [B, S, 64]
  const float* w = (const float*)d_in[1];  // [64, 64]
  const float* b = (const float*)d_in[2];  // [64]
  float* y = (float*)d_out;                // [B, S, 64]

  // 1024 blocks x 8 waves = 8192 waves -> 16 tiles per wave (amortizes the
  // one-time 128-VGPR weight preload).
  dim3 grid(1024), block(256);
  lin64_wmma_f32<<<grid, block, 0, stream>>>(x, w, b, y);
}